// LatentMatcher_22720376996613
// MI455X (gfx1250) — compile-verified
//
#include <hip/hip_runtime.h>
#include <math.h>

// Problem constants (match reference)
#define B_  32
#define Q_  512
#define G_  100
#define D_  1024
#define MT_ 7        // ceil(G/16) GT tiles
#define PAD_COST_ 1000.0f
#define INF_ 1e9f

typedef __attribute__((ext_vector_type(2))) float v2f;
typedef __attribute__((ext_vector_type(8))) float v8f;

// ---------------------------------------------------------------------------
// Kernel A: cost[b][g][q] = pad[b][g] ? (1 - gt[b,g,:].q[b,q,:] + 1 - sigmoid(seed[b,q]))
//                                     : PAD_COST
// One wave per 16x16 (g,q) tile, fp32 WMMA 16x16x4 over D=1024 (256 steps).
// A-matrix (16x4 f32): lane%16 = M row; lanes<16 hold K=base+{0,1}, lanes>=16 K=base+{2,3}.
// B-matrix (4x16 f32): lane%16 = N col; same K split. Both -> contiguous float2 loads.
// C/D (16x16 f32, 8 VGPRs): VGPR r -> M = r + (lane<16 ? 0 : 8), N = lane%16.
// ---------------------------------------------------------------------------
__global__ __launch_bounds__(256)
void cost_wmma_kernel(const float* __restrict__ q,     // (B,Q,D)
                      const float* __restrict__ gt,    // (B,G,D)
                      const int*   __restrict__ pad,   // (B,G) nonzero = valid
                      const float* __restrict__ seed,  // (B,Q)
                      float* __restrict__ cost)        // (B,G,Q)
{
    const int lane  = threadIdx.x & 31;
    const int wave  = threadIdx.x >> 5;           // 0..7
    const int ntile = blockIdx.x * 8 + wave;      // 0..31 (Q tiles)
    const int mtile = blockIdx.y;                 // 0..6  (G tiles)
    const int b     = blockIdx.z;

    const int half = lane >> 4;                   // 0: K+{0,1}, 1: K+{2,3}
    const int l16  = lane & 15;

    int arow = mtile * 16 + l16;                  // GT row for A operand
    if (arow >= G_) arow = G_ - 1;                // clamp OOB loads (stores are guarded)
    const int bcol = ntile * 16 + l16;            // query column for B operand

    const float* ap = gt + ((size_t)b * G_ + arow) * D_ + half * 2;
    const float* bp = q  + ((size_t)b * Q_ + bcol) * D_ + half * 2;

    v8f acc = {};
    for (int k = 0; k < D_; k += 4) {
        v2f a  = *(const v2f*)(ap + k);
        v2f bb = *(const v2f*)(bp + k);
        acc = __builtin_amdgcn_wmma_f32_16x16x4_f32(
            /*neg_a=*/false, a, /*neg_b=*/false, bb,
            /*c_mod=*/(short)0, acc, /*reuse_a=*/false, /*reuse_b=*/false);
    }

    // per-query seed cost: 1 - sigmoid(seed)
    const float s = seed[(size_t)b * Q_ + bcol];
    const float seed_cost = 1.0f - 1.0f / (1.0f + expf(-s));

    #pragma unroll
    for (int r = 0; r < 8; ++r) {
        const int row = mtile * 16 + r + half * 8;
        if (row < G_) {
            const float cv = (pad[b * G_ + row] != 0)
                           ? (1.0f - acc[r] + seed_cost)
                           : PAD_COST_;
            cost[((size_t)b * G_ + row) * Q_ + bcol] = cv;
        }
    }
}

// ---------------------------------------------------------------------------
// Kernel B: Jonker-Volgenant shortest-augmenting-path assignment, one block
// per batch. 512 threads; thread t owns column j = t+1 (v, minv, used live in
// registers). Argmin = wave32 shuffle reduce + 1 LDS cross-wave stage, with
// lowest-index tie-break to match jnp.argmin.
// ---------------------------------------------------------------------------
__global__ __launch_bounds__(512)
void hungarian_kernel(const float* __restrict__ cost, // (B,G,Q)
                      const int*   __restrict__ pad,  // (B,G)
                      float* __restrict__ out_mask,   // (B,Q) 0/1
                      float* __restrict__ out_idx)    // (B,Q) gt idx or -1
{
    const int b = blockIdx.x;
    const int t = threadIdx.x;        // 0..511
    const int j = t + 1;              // owned column (1-based)
    const float* C = cost + (size_t)b * G_ * Q_;

    __shared__ float u[G_ + 1];       // row potentials (1-based)
    __shared__ int   p[Q_ + 1];       // p[j]: row matched to col j (0 = free)
    __shared__ int   s_way[Q_ + 1];   // predecessor col on shortest path
    __shared__ float s_rval[16];
    __shared__ int   s_ridx[16];
    __shared__ float s_delta;
    __shared__ int   s_j1;
    __shared__ int   s_j0;

    if (t <= G_) u[t] = 0.0f;
    p[j] = 0;
    if (t == 0) p[0] = 0;
    float v_j = 0.0f;                 // column potential (register)
    __syncthreads();

    for (int i = 1; i <= G_; ++i) {
        if (t == 0) { p[0] = i; s_j0 = 0; }
        float minv_j = INF_;
        int   used_j = 0;
        __syncthreads();

        for (int step = 0; step <= Q_ + 1; ++step) {
            const int j0 = s_j0;
            const int i0 = p[j0];
            if (j == j0) used_j = 1;              // mark current column used
            const float u_i0 = u[i0];

            // relax: reduced cost of row i0 into my column
            float myval = INF_;
            if (!used_j) {
                const float cur = C[(size_t)(i0 - 1) * Q_ + t] - u_i0 - v_j;
                if (cur < minv_j) { minv_j = cur; s_way[j] = j0; }
                myval = minv_j;
            }
            int myidx = j;

            // wave32 argmin (ties -> lower column index)
            #pragma unroll
            for (int off = 16; off > 0; off >>= 1) {
                const float ov = __shfl_xor(myval, off, 32);
                const int   oi = __shfl_xor(myidx, off, 32);
                if (ov < myval || (ov == myval && oi < myidx)) { myval = ov; myidx = oi; }
            }
            const int wid = t >> 5;
            if ((t & 31) == 0) { s_rval[wid] = myval; s_ridx[wid] = myidx; }
            __syncthreads();

            if (t < 32) {                          // wave 0 combines 16 partials
                float rv = (t < 16) ? s_rval[t] : INF_;
                int   ri = (t < 16) ? s_ridx[t] : 0x7fffffff;
                #pragma unroll
                for (int off = 8; off > 0; off >>= 1) {
                    const float ov = __shfl_xor(rv, off, 32);
                    const int   oi = __shfl_xor(ri, off, 32);
                    if (ov < rv || (ov == rv && oi < ri)) { rv = ov; ri = oi; }
                }
                if (t == 0) { s_delta = rv; s_j1 = ri; }
            }
            __syncthreads();

            const float delta = s_delta;
            const int   j1    = s_j1;

            // potential updates (used columns map to distinct rows -> race-free)
            if (used_j) {
                u[p[j]] += delta;
                v_j     -= delta;
            } else {
                minv_j  -= delta;
            }
            if (t == 0) {
                u[i] += delta;                     // virtual column 0 (p[0]=i, always used)
                s_j0 = j1;
            }
            __syncthreads();

            if (p[j1] == 0) break;                 // reached a free column -> augment
        }

        if (t == 0) {                              // backtrack augmenting path
            int jj = s_j0;
            while (jj != 0) {
                const int jn = s_way[jj];
                p[jj] = p[jn];
                jj = jn;
            }
        }
        __syncthreads();
    }

    // emit: mask + matched gt index per query column
    const int row = p[j] - 1;
    const int ok  = (row >= 0) && (pad[b * G_ + row] != 0);
    out_mask[(size_t)b * Q_ + t] = ok ? 1.0f : 0.0f;
    out_idx [(size_t)b * Q_ + t] = ok ? (float)row : -1.0f;
}

// ---------------------------------------------------------------------------
extern "C" void kernel_launch(void* const* d_in, const int* in_sizes, int n_in,
                              void* d_out, int out_size, void* d_ws, size_t ws_size,
                              hipStream_t stream) {
    (void)in_sizes; (void)n_in; (void)out_size; (void)ws_size;

    const float* q    = (const float*)d_in[0];   // (B,Q,D) f32
    const float* gt   = (const float*)d_in[1];   // (B,G,D) f32 (unit rows)
    const int*   pad  = (const int*)d_in[2];     // (B,G) bool->int
    const float* seed = (const float*)d_in[3];   // (B,Q) f32

    float* cost     = (float*)d_ws;              // (B,G,Q) = 6.25 MiB scratch
    float* out_mask = (float*)d_out;             // (B,Q)
    float* out_idx  = out_mask + (size_t)B_ * Q_;// (B,Q)

    dim3 gridA(4, MT_, B_);                      // 4 blocks x 8 waves = 32 Q-tiles
    cost_wmma_kernel<<<gridA, dim3(256), 0, stream>>>(q, gt, pad, seed, cost);

    hungarian_kernel<<<dim3(B_), dim3(Q_), 0, stream>>>(cost, pad, out_mask, out_idx);
}